// LearnableLaplacianBasis_1245540515921
// MI455X (gfx1250) — compile-verified
//
#include <hip/hip_runtime.h>
#include <math.h>

typedef __attribute__((ext_vector_type(2))) float v2f;
typedef __attribute__((ext_vector_type(8))) float v8f;

#define B_  16
#define T_  4096
#define D_  512
#define K_  128
#define BT_ (B_*T_)
#define CH_ 64              // chunks along T
#define L_  (T_/CH_)        // steps per chunk = 64
#define DC_ 32              // D-chunk staged through LDS
#define NCHUNK_ (D_/DC_)    // 16
#define PAD_ 36             // padded LDS row stride (floats): gcd(36,64)=4 -> conflict-free column reads

#define ALPHA_MIN_ 1e-6f
#define OMEGA_MAX_ 3.14159265358979323846f
#define SN_EPS_    1e-6f

// workspace layout (float offsets)
static constexpr size_t WS_INV_SIGMA = 0;
static constexpr size_t WS_ALPHA     = 16;
static constexpr size_t WS_OMEGA     = WS_ALPHA + K_;
static constexpr size_t WS_BGAIN     = WS_OMEGA + K_;
static constexpr size_t WS_INJ       = 512;                                   // [BT_,K_]   32 MB
static constexpr size_t WS_SUM       = WS_INJ + (size_t)BT_*K_;               // [B,CH,K,4]  2 MB
static constexpr size_t WS_PRE       = WS_SUM + (size_t)B_*CH_*K_*4;          // [B,CH,K,2]  1 MB

// ---- CDNA5 async global->LDS staging (guarded; falls back to sync staging) ----
#if __has_builtin(__builtin_amdgcn_global_load_async_to_lds_b128)
#define ASYNC_LDS 1
#else
#define ASYNC_LDS 0
#endif

#if ASYNC_LDS
typedef int v4i_ __attribute__((vector_size(4 * sizeof(int))));
typedef __attribute__((address_space(1))) v4i_* gas_t;   // global, b128 pointee
typedef __attribute__((address_space(3))) v4i_* las_t;   // LDS,    b128 pointee
__device__ __forceinline__ gas_t as_g(const void* p) {
  return (gas_t)(unsigned long long)p;                   // VA-preserving int round-trip
}
__device__ __forceinline__ las_t as_l(const void* p) {
  // generic LDS address: low 32 bits are the LDS byte offset (ISA aperture rule)
  return (las_t)(unsigned)(unsigned long long)p;
}
#if __has_builtin(__builtin_amdgcn_s_wait_asynccnt)
#define WAIT_ASYNC(n) __builtin_amdgcn_s_wait_asynccnt(n)
#else
#define WAIT_ASYNC(n) asm volatile("s_wait_asynccnt %0" :: "i"(n) : "memory")
#endif
#else
#define WAIT_ASYNC(n)
#endif

// ---------------------------------------------------------------------------
// Kernel 1: spectral-norm power iteration + per-k coefficient precompute
// ---------------------------------------------------------------------------
__global__ __launch_bounds__(256)
void lap_setup(const float* __restrict__ W, const float* __restrict__ u_sn,
               const float* __restrict__ s_real_raw, const float* __restrict__ s_imag,
               const float* __restrict__ b_log, float* __restrict__ ws) {
  __shared__ float v[D_];
  __shared__ float red[256];
  __shared__ float u2[K_];
  __shared__ float sc;
  const int tid = threadIdx.x;

  // v = W^T u
  for (int d = tid; d < D_; d += 256) {
    float acc = 0.f;
    for (int k = 0; k < K_; ++k) acc += W[(size_t)k*D_ + d] * u_sn[k];
    v[d] = acc;
  }
  __syncthreads();
  // v /= (||v|| + eps)
  float s = 0.f;
  for (int d = tid; d < D_; d += 256) s += v[d]*v[d];
  red[tid] = s; __syncthreads();
  for (int off = 128; off > 0; off >>= 1) { if (tid < off) red[tid] += red[tid+off]; __syncthreads(); }
  if (tid == 0) sc = 1.f / (sqrtf(red[0]) + SN_EPS_);
  __syncthreads();
  const float vin = sc;
  for (int d = tid; d < D_; d += 256) v[d] *= vin;
  __syncthreads();
  // u2_un = W v ; sigma = ||u2_un||^2 / (||u2_un|| + eps)
  if (tid < K_) {
    float acc = 0.f;
    for (int d = 0; d < D_; ++d) acc += W[(size_t)tid*D_ + d] * v[d];
    u2[tid] = acc;
  }
  __syncthreads();
  red[tid] = (tid < K_) ? u2[tid]*u2[tid] : 0.f;
  __syncthreads();
  for (int off = 128; off > 0; off >>= 1) { if (tid < off) red[tid] += red[tid+off]; __syncthreads(); }
  if (tid == 0) {
    float nu2 = red[0];
    float nu  = sqrtf(nu2);
    float sigma = nu2 / (nu + SN_EPS_);
    ws[WS_INV_SIGMA] = 1.f / sigma;
  }
  if (tid < K_) {
    float xr = s_real_raw[tid];
    float sp = (xr > 20.f) ? xr : log1pf(expf(xr));          // softplus
    ws[WS_ALPHA + tid] = sp + ALPHA_MIN_;
    float om = s_imag[tid];
    ws[WS_OMEGA + tid] = fminf(fmaxf(om, -OMEGA_MAX_), OMEGA_MAX_);
    ws[WS_BGAIN + tid] = expf(b_log[tid]);
  }
}

// ---------------------------------------------------------------------------
// Stage one (16 x DC_) X tile + (128 x DC_) W chunk into (padded) LDS buffers.
// Async path: global_load_async_to_lds_b128, tracked with ASYNCcnt.
// ---------------------------------------------------------------------------
__device__ __forceinline__
void stage_chunk(const float* __restrict__ x, const float* __restrict__ W,
                 float* Xb, float* Wb, int tid, int m0, int d0) {
#if ASYNC_LDS
  if (tid < 128) {                       // X: 16 rows x 8 float4
    int row = tid >> 3, col = (tid & 7) * 4;
    __builtin_amdgcn_global_load_async_to_lds_b128(
        as_g(&x[(size_t)(m0 + row)*D_ + d0 + col]),
        as_l(&Xb[row*PAD_ + col]), 0, 0);
  }
  for (int i = 0; i < 4; ++i) {          // W: 128 rows x 8 float4 = 1024
    int lin = tid + i*256;
    int row = lin >> 3, col = (lin & 7) * 4;
    __builtin_amdgcn_global_load_async_to_lds_b128(
        as_g(&W[(size_t)row*D_ + d0 + col]),
        as_l(&Wb[row*PAD_ + col]), 0, 0);
  }
#else
  if (tid < 128) {
    int row = tid >> 3, col = (tid & 7) * 4;
    float4 v = *(const float4*)&x[(size_t)(m0 + row)*D_ + d0 + col];
    float* d = &Xb[row*PAD_ + col];
    d[0]=v.x; d[1]=v.y; d[2]=v.z; d[3]=v.w;
  }
  for (int i = 0; i < 4; ++i) {
    int lin = tid + i*256;
    int row = lin >> 3, col = (lin & 7) * 4;
    float4 v = *(const float4*)&W[(size_t)row*D_ + d0 + col];
    float* d = &Wb[row*PAD_ + col];
    d[0]=v.x; d[1]=v.y; d[2]=v.z; d[3]=v.w;
  }
#endif
}

// ---------------------------------------------------------------------------
// Kernel 2: u = x @ (W/sigma)^T + bias, fused epilogue -> inj = (1-rho)*b_gain*u
// fp32 WMMA (16x16x4), 8 waves/block, M-tile=16 rows, wave w owns N-tile [16w,16w+16)
// Double-buffered LDS; async DMA pipelined one chunk ahead via ASYNCcnt.
// ---------------------------------------------------------------------------
__global__ __launch_bounds__(256)
void lap_proj(const float* __restrict__ x, const float* __restrict__ dt,
              const float* __restrict__ W, const float* __restrict__ bias,
              float* __restrict__ ws) {
  __shared__ float Xs [2][16 * PAD_];    // 2 x  16 x 32 (+pad)  ~4.5 KB
  __shared__ float Wsh[2][K_ * PAD_];    // 2 x 128 x 32 (+pad)  ~36 KB

  const int tid  = threadIdx.x;
  const int lane = tid & 31;
  const int wave = tid >> 5;             // 0..7 -> N tile base = wave*16
  const int m0   = blockIdx.x * 16;      // global row tile (flattened B*T)
  const float inv_sigma = ws[WS_INV_SIGMA];

  v8f acc = {0.f,0.f,0.f,0.f,0.f,0.f,0.f,0.f};

  const int m    = lane & 15;            // A-fragment row
  const int koff = (lane >> 4) * 2;      // A/B fragment K sub-offset
  const int n    = (wave << 4) + (lane & 15);

  stage_chunk(x, W, Xs[0], Wsh[0], tid, m0, 0);
  for (int c = 0; c < NCHUNK_; ++c) {
    if (c + 1 < NCHUNK_) {
      stage_chunk(x, W, Xs[(c+1)&1], Wsh[(c+1)&1], tid, m0, (c+1)*DC_);
      WAIT_ASYNC(4);                     // drain chunk c; keep chunk c+1 in flight
    } else {
      WAIT_ASYNC(0);
    }
    __syncthreads();                     // all waves' DMA for chunk c visible
    const float* Xb = Xs[c&1];
    const float* Wb = Wsh[c&1];
#pragma unroll
    for (int dk = 0; dk < DC_; dk += 4) {
      v2f a = *(const v2f*)&Xb[m*PAD_ + dk + koff];
      v2f b = *(const v2f*)&Wb[n*PAD_ + dk + koff];
      acc = __builtin_amdgcn_wmma_f32_16x16x4_f32(false, a, false, b,
                                                  (short)0, acc, false, false);
    }
    __syncthreads();                     // done reading buf c&1 -> safe to overwrite
  }

  // epilogue: sigma scale + bias, then inj = (1 - exp(-alpha*dt)) * b_gain * u
  const float bi = bias[n];
  const float al = ws[WS_ALPHA + n];
  const float bg = ws[WS_BGAIN + n];
  const int mhalf = (lane >> 4) * 8;     // C/D layout: vgpr g -> row g + 8*(lane>=16)
  for (int g = 0; g < 8; ++g) {
    int r = m0 + mhalf + g;              // flattened (b,t) row
    float u = acc[g] * inv_sigma + bi;
    float rho = expf(-al * dt[r]);
    ws[WS_INJ + (size_t)r*K_ + n] = (1.f - rho) * bg * u;
  }
}

// ---------------------------------------------------------------------------
// Scan phase 1: per-chunk affine composition  z -> M*z + Dv  (M,Dv complex)
// ---------------------------------------------------------------------------
__global__ __launch_bounds__(K_)
void lap_scan_ph1(const float* __restrict__ dt, float* __restrict__ ws) {
  const int k = threadIdx.x, c = blockIdx.x, b = blockIdx.y;
  const float al = ws[WS_ALPHA + k];
  const float om = ws[WS_OMEGA + k];
  const float* __restrict__ inj = ws + WS_INJ;
  float Mr = 1.f, Mi = 0.f, Dx = 0.f, Dy = 0.f;
  const int t0 = c * L_;
  for (int j = 0; j < L_; ++j) {
    const int t = t0 + j;
    const float dtv = dt[b*T_ + t];
    const float r  = expf(-al * dtv);
    const float th = om * dtv;
    const float ct = cosf(th), st = sinf(th);
    const float mr = r*ct, mi = r*st;
    const float it = inj[((size_t)b*T_ + t)*K_ + k];
    float nMr = mr*Mr - mi*Mi;
    float nMi = mr*Mi + mi*Mr;
    float nDx = mr*Dx - mi*Dy + it;
    float nDy = mr*Dy + mi*Dx + it;
    Mr = nMr; Mi = nMi; Dx = nDx; Dy = nDy;
  }
  const size_t o = WS_SUM + ((size_t)(b*CH_ + c)*K_ + k)*4;
  ws[o] = Mr; ws[o+1] = Mi; ws[o+2] = Dx; ws[o+3] = Dy;
}

// ---------------------------------------------------------------------------
// Scan phase 2: exclusive scan over the 64 chunk summaries per (b,k)
// ---------------------------------------------------------------------------
__global__ __launch_bounds__(K_)
void lap_scan_ph2(float* __restrict__ ws) {
  const int k = threadIdx.x, b = blockIdx.x;
  float zr = 0.f, zi = 0.f;
  for (int c = 0; c < CH_; ++c) {
    const size_t p = WS_PRE + ((size_t)(b*CH_ + c)*K_ + k)*2;
    ws[p] = zr; ws[p+1] = zi;                          // state entering chunk c
    const size_t o = WS_SUM + ((size_t)(b*CH_ + c)*K_ + k)*4;
    const float Mr = ws[o], Mi = ws[o+1], Dx = ws[o+2], Dy = ws[o+3];
    float nr = Mr*zr - Mi*zi + Dx;
    float ni = Mr*zi + Mi*zr + Dy;
    zr = nr; zi = ni;
  }
}

// ---------------------------------------------------------------------------
// Scan phase 3: replay within each chunk from its prefix state, write C|S
// (streaming output -> non-temporal stores, keep inj resident in L2 instead)
// ---------------------------------------------------------------------------
__global__ __launch_bounds__(K_)
void lap_scan_ph3(const float* __restrict__ dt, const float* __restrict__ ws,
                  float* __restrict__ out) {
  const int k = threadIdx.x, c = blockIdx.x, b = blockIdx.y;
  const float al = ws[WS_ALPHA + k];
  const float om = ws[WS_OMEGA + k];
  const float* __restrict__ inj = ws + WS_INJ;
  const size_t p = WS_PRE + ((size_t)(b*CH_ + c)*K_ + k)*2;
  float zr = ws[p], zi = ws[p+1];
  const int t0 = c * L_;
  for (int j = 0; j < L_; ++j) {
    const int t = t0 + j;
    const float dtv = dt[b*T_ + t];
    const float r  = expf(-al * dtv);
    const float th = om * dtv;
    const float ct = cosf(th), st = sinf(th);
    const float mr = r*ct, mi = r*st;
    const float it = inj[((size_t)b*T_ + t)*K_ + k];
    float nr = mr*zr - mi*zi + it;     // c_new = r*(c*ct - s*st) + it
    float ni = mi*zr + mr*zi + it;     // s_new = r*(c*st + s*ct) + it
    zr = nr; zi = ni;
    const size_t ob = ((size_t)b*T_ + t)*(2*K_);
    __builtin_nontemporal_store(zr, &out[ob + k]);
    __builtin_nontemporal_store(zi, &out[ob + K_ + k]);
  }
}

// ---------------------------------------------------------------------------
extern "C" void kernel_launch(void* const* d_in, const int* in_sizes, int n_in,
                              void* d_out, int out_size, void* d_ws, size_t ws_size,
                              hipStream_t stream) {
  const float* x    = (const float*)d_in[0];
  const float* dt   = (const float*)d_in[1];
  const float* sr   = (const float*)d_in[2];
  const float* si   = (const float*)d_in[3];
  const float* W    = (const float*)d_in[4];
  const float* bias = (const float*)d_in[5];
  const float* blog = (const float*)d_in[6];
  const float* usn  = (const float*)d_in[7];
  float* out = (float*)d_out;
  float* ws  = (float*)d_ws;

  lap_setup<<<1, 256, 0, stream>>>(W, usn, sr, si, blog, ws);
  lap_proj<<<BT_/16, 256, 0, stream>>>(x, dt, W, bias, ws);
  lap_scan_ph1<<<dim3(CH_, B_), K_, 0, stream>>>(dt, ws);
  lap_scan_ph2<<<B_, K_, 0, stream>>>(ws);
  lap_scan_ph3<<<dim3(CH_, B_), K_, 0, stream>>>(dt, ws, out);
}